// PowerSVDLayer_87522843558173
// MI455X (gfx1250) — compile-verified
//
#include <hip/hip_runtime.h>
#include <hip/hip_bf16.h>

typedef __attribute__((ext_vector_type(2))) float v2f;
typedef __attribute__((ext_vector_type(8))) float v8f;
typedef __attribute__((ext_vector_type(4))) unsigned int v4u;
typedef __attribute__((ext_vector_type(8))) unsigned int v8u;

#define MDIM 256
#define NDIM 256
#define RR   8
#define ASTR 257   // A row stride in LDS (floats): conflict-free column reads
#define PSTR 16    // P/Q row stride: cols 8..15 kept zero -> unconditional B loads

// Cholesky-QR of an LDS-resident 256x8 matrix (stored [256][PSTR]), in place.
__device__ __forceinline__ void cholqr256x8(float* __restrict__ Mx,
                                            float* __restrict__ G,
                                            float* __restrict__ L,
                                            int t) {
  __syncthreads();
  // Gram matrix G = M^T M  (waves 0-1; one entry per thread)
  if (t < 64) {
    const int i = t >> 3, j = t & 7;
    float s = 0.f;
    #pragma unroll 8
    for (int r = 0; r < 256; ++r)
      s += Mx[r * PSTR + i] * Mx[r * PSTR + j];
    G[t] = s;
  }
  __syncthreads();
  // 8x8 Cholesky, single thread; store reciprocal on the diagonal
  if (t == 0) {
    float Lm[8][8];
    for (int i = 0; i < 8; ++i)
      for (int j = 0; j < 8; ++j) Lm[i][j] = 0.f;
    for (int i = 0; i < 8; ++i) {
      float d = G[i * 8 + i];
      for (int k = 0; k < i; ++k) d -= Lm[i][k] * Lm[i][k];
      d = sqrtf(fmaxf(d, 1e-30f));
      const float inv = 1.0f / d;
      Lm[i][i] = d;
      for (int j = i + 1; j < 8; ++j) {
        float s = G[j * 8 + i];
        for (int k = 0; k < i; ++k) s -= Lm[j][k] * Lm[i][k];
        Lm[j][i] = s * inv;
      }
      Lm[i][i] = inv;             // keep 1/d for the substitution pass
    }
    for (int i = 0; i < 8; ++i)
      for (int j = 0; j < 8; ++j) L[i * 8 + j] = Lm[i][j];
  }
  __syncthreads();
  // Q = M * L^-T : per-row forward substitution (thread t owns row t)
  float r[8], q[8];
  #pragma unroll
  for (int j = 0; j < 8; ++j) r[j] = Mx[t * PSTR + j];
  #pragma unroll
  for (int j = 0; j < 8; ++j) {
    float s = r[j];
    for (int i = 0; i < j; ++i) s -= q[i] * L[j * 8 + i];
    q[j] = s * L[j * 8 + j];      // diagonal holds reciprocal
  }
  #pragma unroll
  for (int j = 0; j < 8; ++j) Mx[t * PSTR + j] = q[j];
  __syncthreads();
}

__global__ __launch_bounds__(256, 1)
void powersvd_wmma_kernel(const float* __restrict__ Ag,
                          const float* __restrict__ P0g,
                          const int*   __restrict__ iterp,
                          float* __restrict__ Og) {
  __shared__ float Ash[MDIM * ASTR];   // 263,168 B : input slice, 257-stride rows
  __shared__ float Psh[NDIM * PSTR];   //  16,384 B : p  [256][16], cols 8..15 == 0
  __shared__ float Qsh[MDIM * PSTR];   //  16,384 B : q  [256][16], cols 8..15 == 0
  __shared__ float Gsh[64];
  __shared__ float Lsh[64];

  const int t    = threadIdx.x;     // 0..255 (8 waves, wave32)
  const int lane = t & 31;
  const int wave = t >> 5;
  const int lr   = lane & 15;       // lane-within-half
  const int hs   = lane >> 4;       // half-wave select (K pairing / M+8)

  const size_t bc = blockIdx.x;
  const float* A  = Ag  + bc * (size_t)(MDIM * NDIM);
  const float* P0 = P0g + bc * (size_t)(NDIM * RR);
  float*       O  = Og  + bc * (size_t)(MDIM * NDIM);
  const int iters = *iterp;

  // ---- TDM: DMA the 256x256 fp32 slice into LDS with 1-DWORD row padding ----
  // D# group0: count=1 | lds_addr | global_addr(57b) | type=2
  // D# group1: data_size=4B, pad_enable, pad_interval=256 DWORDs, pad_amount=1 DWORD,
  //            tensor 256x256, tile 256x256, dim0_stride=256
  if (wave == 0) {
    const unsigned long long ga = (unsigned long long)(uintptr_t)A;
    const unsigned lds = (unsigned)(uintptr_t)(&Ash[0]);
    v4u g0;
    g0[0] = 1u;                                               // count=1
    g0[1] = lds;                                              // lds_addr
    g0[2] = (unsigned)(ga & 0xFFFFFFFFull);                   // global_addr[31:0]
    g0[3] = (unsigned)((ga >> 32) & 0x1FFFFFFull)             // global_addr[56:32]
            | (2u << 30);                                     // type=2 ("image")
    v8u g1;
    g1[0] = (2u << 16) | (1u << 20) | (7u << 22);             // 4B, pad_en, ivl=256dw, amt=1dw
    g1[1] = (256u & 0xFFFFu) << 16;                           // tensor_dim0 = 256 (lo)
    g1[2] = 256u << 16;                                       // dim0 hi=0 | tensor_dim1=256 (lo)
    g1[3] = 256u << 16;                                       // dim1 hi=0 | tile_dim0=256
    g1[4] = 256u;                                             // tile_dim1=256, tile_dim2=0
    g1[5] = 256u;                                             // tensor_dim0_stride=256 (lo)
    g1[6] = 0u;                                               // stride hi / dim1_stride lo
    g1[7] = 0u;
    asm volatile("tensor_load_to_lds %0, %1" :: "s"(g0), "s"(g1) : "memory");
    __builtin_amdgcn_s_wait_tensorcnt(0);
  }

  // p init: cols 0..7 from p0, cols 8..15 zero (invariant kept by full-tile stores)
  for (int i = t; i < NDIM * PSTR; i += 256) {
    const int r = i >> 4, c = i & 15;
    Psh[i] = (c < RR) ? P0[r * RR + c] : 0.f;
  }
  __syncthreads();

  for (int it = 0; it < iters; ++it) {
    // p = qr(p).Q
    cholqr256x8(Psh, Gsh, Lsh, t);

    // q = A @ p   (M=256,N=8->16,K=256), v_wmma_f32_16x16x4_f32, 2 M-tiles/wave
    #pragma unroll
    for (int tt = 0; tt < 2; ++tt) {
      const int m0 = (wave * 2 + tt) * 16;
      v8f acc = {0.f, 0.f, 0.f, 0.f, 0.f, 0.f, 0.f, 0.f};
      for (int k = 0; k < NDIM; k += 4) {
        const int kk = k + 2 * hs;
        v2f a, b;
        a[0] = Ash[(m0 + lr) * ASTR + kk];
        a[1] = Ash[(m0 + lr) * ASTR + kk + 1];
        b[0] = Psh[kk * PSTR + lr];
        b[1] = Psh[(kk + 1) * PSTR + lr];
        acc = __builtin_amdgcn_wmma_f32_16x16x4_f32(false, a, false, b,
                                                    (short)0, acc, false, false);
      }
      #pragma unroll
      for (int v = 0; v < 8; ++v)            // cols 8..15 of acc are exactly 0
        Qsh[(m0 + v + 8 * hs) * PSTR + lr] = acc[v];
    }
    __syncthreads();

    // q = qr(q).Q
    cholqr256x8(Qsh, Gsh, Lsh, t);

    // p = A^T @ q  (column reads of Ash conflict-free via 257 stride)
    #pragma unroll
    for (int tt = 0; tt < 2; ++tt) {
      const int n0 = (wave * 2 + tt) * 16;
      v8f acc = {0.f, 0.f, 0.f, 0.f, 0.f, 0.f, 0.f, 0.f};
      for (int k = 0; k < MDIM; k += 4) {
        const int kk = k + 2 * hs;
        v2f a, b;
        a[0] = Ash[kk * ASTR + n0 + lr];         // A^T row = A column
        a[1] = Ash[(kk + 1) * ASTR + n0 + lr];
        b[0] = Qsh[kk * PSTR + lr];
        b[1] = Qsh[(kk + 1) * PSTR + lr];
        acc = __builtin_amdgcn_wmma_f32_16x16x4_f32(false, a, false, b,
                                                    (short)0, acc, false, false);
      }
      #pragma unroll
      for (int v = 0; v < 8; ++v)
        Psh[(n0 + v + 8 * hs) * PSTR + lr] = acc[v];
    }
    __syncthreads();
  }

  // out = q @ p^T  (M=256,N=256,K=8): 256 tiles, 32 per wave, 2 WMMA per tile
  for (int tile = wave; tile < 256; tile += 8) {
    const int m0 = (tile >> 4) * 16;
    const int n0 = (tile & 15) * 16;
    v8f acc = {0.f, 0.f, 0.f, 0.f, 0.f, 0.f, 0.f, 0.f};
    #pragma unroll
    for (int k = 0; k < RR; k += 4) {
      const int kk = k + 2 * hs;
      v2f a, b;
      a[0] = Qsh[(m0 + lr) * PSTR + kk];
      a[1] = Qsh[(m0 + lr) * PSTR + kk + 1];
      b[0] = Psh[(n0 + lr) * PSTR + kk];       // (p^T)[k][n] = p[n][k]
      b[1] = Psh[(n0 + lr) * PSTR + kk + 1];
      acc = __builtin_amdgcn_wmma_f32_16x16x4_f32(false, a, false, b,
                                                  (short)0, acc, false, false);
    }
    #pragma unroll
    for (int v = 0; v < 8; ++v)
      O[(size_t)(m0 + v + 8 * hs) * NDIM + n0 + lr] = acc[v];
  }
}

extern "C" void kernel_launch(void* const* d_in, const int* in_sizes, int n_in,
                              void* d_out, int out_size, void* d_ws, size_t ws_size,
                              hipStream_t stream) {
  const float* input = (const float*)d_in[0];  // [B,C,256,256] fp32
  const float* p0    = (const float*)d_in[1];  // [B,C,256,8]   fp32
  // d_in[2] = q0 (overwritten before use in the reference)
  const int*   iter  = (const int*)d_in[3];    // scalar int (==2)
  float* out = (float*)d_out;                  // [B,C,256,256] fp32

  const int bcCount = in_sizes[0] / (MDIM * NDIM);  // B*C = 512
  powersvd_wmma_kernel<<<dim3(bcCount), dim3(256), 0, stream>>>(input, p0, iter, out);
}